// OscillatorEngine_37752762532595
// MI455X (gfx1250) — compile-verified
//
#include <hip/hip_runtime.h>

#define NCELLS 8192
#define TWO_PI_F 6.283185307179586f

typedef __attribute__((ext_vector_type(2))) float v2f;
typedef __attribute__((ext_vector_type(8))) float v8f;

// ---------------------------------------------------------------------------
// K0: ||x||, sin/cos of original phases, pack padded A-panel
//     Ppad[16][1536] = [w_p2h0|0 ; w_p2h1|0 ; b_p2h|x ; 0...], zero wpad rows 1..15
// ---------------------------------------------------------------------------
__global__ __launch_bounds__(256) void prep_kernel(
    const float* __restrict__ x, const float* __restrict__ phases,
    const float* __restrict__ w_p2h, const float* __restrict__ b_p2h,
    float* __restrict__ sin_o, float* __restrict__ cos_o,
    float* __restrict__ scal, float* __restrict__ Ppad,
    float* __restrict__ wpad)
{
    __shared__ float red[256];
    const int t = threadIdx.x;
    float s = 0.f;
    for (int i = t; i < 512; i += 256) { float v = x[i]; s += v * v; }
    red[t] = s; __syncthreads();
    for (int off = 128; off > 0; off >>= 1) {
        if (t < off) red[t] += red[t + off];
        __syncthreads();
    }
    if (t == 0) scal[0] = sqrtf(red[0]);

    for (int i = t; i < NCELLS; i += 256) {
        float p = phases[i];
        sin_o[i] = sinf(p);
        cos_o[i] = cosf(p);
    }
    for (int idx = t; idx < 16 * 1536; idx += 256) {
        int r = idx / 1536, d = idx - r * 1536;
        float v = 0.f;
        if (r == 0)      v = (d < 1024) ? w_p2h[d] : 0.f;
        else if (r == 1) v = (d < 1024) ? w_p2h[1024 + d] : 0.f;
        else if (r == 2) v = (d < 1024) ? b_p2h[d] : x[d - 1024];
        Ppad[idx] = v;
    }
    for (int idx = t; idx < 15 * 1024; idx += 256)
        wpad[1024 + idx] = 0.f;   // rows 1..15 of weighted panel
}

// ---------------------------------------------------------------------------
// Thin GEMM via V_WMMA_F32_16X16X4_F32 (uniform control flow, EXEC all-ones):
//   out[r][col] = sum_d P[r*kdim+d] * W[d*ncols+col], r < orows (<=3)
//   P is a FULLY-INITIALIZED 16 x kdim panel (dead rows are zero).
// grid = ncols/16 blocks, 256 threads: 8 waves split K, LDS reduce.
// A layout (ISA 7.12.2, f32 16x4): lane&15 = M; VGPR0 = K(half?2:0), VGPR1 = K+1
// B layout (mirror):               lane&15 = N; VGPR0 = K(half?2:0), VGPR1 = K+1
// ---------------------------------------------------------------------------
__global__ __launch_bounds__(256) void rank_gemm_wmma(
    const float* __restrict__ P, int kdim,
    const float* __restrict__ W, int ncols,
    const float* __restrict__ bias, int biasRow,
    float* __restrict__ out, int orows)
{
    const int col0 = blockIdx.x * 16;
    const int lane = threadIdx.x & 31;
    const int wave = threadIdx.x >> 5;     // 0..7
    const int half = lane >> 4;
    const int m    = lane & 15;            // A row / B col
    const int kchunk = kdim >> 3;          // per-wave K span (multiple of 8)
    const int kbeg = wave * kchunk;
    const int koff = half ? 2 : 0;

    const float* pa = P + (size_t)m * kdim + kbeg + koff;            // even offset: b64-able
    const float* wp = W + (size_t)(kbeg + koff) * ncols + col0 + m;
    const int nIter2 = kchunk >> 3;        // pairs of K=4 steps

    v8f c0 = {0.f,0.f,0.f,0.f,0.f,0.f,0.f,0.f};
    v8f c1 = {0.f,0.f,0.f,0.f,0.f,0.f,0.f,0.f};
    const int bstride = 4 * ncols;
    for (int it = 0; it < nIter2; ++it) {
        v2f a0 = *(const v2f*)pa;          // K = kA, kA+1
        v2f a1 = *(const v2f*)(pa + 4);
        v2f b0, b1;
        b0.x = wp[0];          b0.y = wp[ncols];
        b1.x = wp[bstride];    b1.y = wp[bstride + ncols];
        pa += 8;
        wp += 2 * bstride;
        c0 = __builtin_amdgcn_wmma_f32_16x16x4_f32(false, a0, false, b0,
                                                   (short)0, c0, false, false);
        c1 = __builtin_amdgcn_wmma_f32_16x16x4_f32(false, a1, false, b1,
                                                   (short)0, c1, false, false);
    }

    // D layout: VGPR r holds M=r (lanes 0-15) / M=8+r (lanes 16-31); need M<orows<=3
    __shared__ float sm[8 * 3 * 16];
    if (lane < 16) {
        sm[(wave * 3 + 0) * 16 + m] = c0[0] + c1[0];
        if (orows > 1) sm[(wave * 3 + 1) * 16 + m] = c0[1] + c1[1];
        if (orows > 2) sm[(wave * 3 + 2) * 16 + m] = c0[2] + c1[2];
    }
    __syncthreads();
    const int t = threadIdx.x;
    if (t < orows * 16) {
        const int r = t >> 4, n = t & 15;
        float s = 0.f;
        for (int w = 0; w < 8; ++w) s += sm[(w * 3 + r) * 16 + n];
        if (bias != nullptr && r == biasRow) s += bias[col0 + n];
        out[(size_t)r * ncols + col0 + n] = s;
    }
}

// ---------------------------------------------------------------------------
// K1: Kuramoto row reduction over the 268 MB coupling stream (HBM-bound).
// One row per wave; sin/cos(phases) staged once per block in 64 KB LDS.
// interactions_i = (K/N)(sin pi * S_j sig(c_ij) cos pj - cos pi * S_j sig(c_ij) sin pj)
// ---------------------------------------------------------------------------
__global__ __launch_bounds__(256) void kuramoto_rows(
    const float* __restrict__ coupling, const float* __restrict__ phases,
    const float* __restrict__ omegas, const float* __restrict__ Kptr,
    const float* __restrict__ sin_o, const float* __restrict__ cos_o,
    const float* __restrict__ scal,
    float* __restrict__ sp, float* __restrict__ cp)
{
    __shared__ __align__(16) float lds_s[NCELLS];
    __shared__ __align__(16) float lds_c[NCELLS];
    for (int j = threadIdx.x; j < NCELLS; j += 256) {
        lds_s[j] = sin_o[j];
        lds_c[j] = cos_o[j];
    }
    __syncthreads();

    const int wave = threadIdx.x >> 5, lane = threadIdx.x & 31;
    const int row = blockIdx.x * 8 + wave;
    const float4* __restrict__ rowp = (const float4*)(coupling + (size_t)row * NCELLS);
    const float4* ls4 = (const float4*)lds_s;
    const float4* lc4 = (const float4*)lds_c;

    float accS = 0.f, accC = 0.f;
    for (int q = lane; q < NCELLS / 4; q += 32) {
        float4 v = rowp[q];
        float4 s4 = ls4[q];
        float4 c4 = lc4[q];
        float g;
        g = 1.0f / (1.0f + __expf(-v.x)); accC = fmaf(g, c4.x, accC); accS = fmaf(g, s4.x, accS);
        g = 1.0f / (1.0f + __expf(-v.y)); accC = fmaf(g, c4.y, accC); accS = fmaf(g, s4.y, accS);
        g = 1.0f / (1.0f + __expf(-v.z)); accC = fmaf(g, c4.z, accC); accS = fmaf(g, s4.z, accS);
        g = 1.0f / (1.0f + __expf(-v.w)); accC = fmaf(g, c4.w, accC); accS = fmaf(g, s4.w, accS);
    }
    for (int off = 16; off > 0; off >>= 1) {
        accS += __shfl_down(accS, off, 32);
        accC += __shfl_down(accC, off, 32);
    }
    if (lane == 0) {
        const float K = Kptr[0];
        const float inter = (K / (float)NCELLS) * (sin_o[row] * accC - cos_o[row] * accS);
        const float dtheta = omegas[row] + inter + 0.1f * scal[0];
        float pn = fmodf(phases[row] + 0.1f * dtheta, TWO_PI_F);
        if (pn < 0.f) pn += TWO_PI_F;   // floored mod, like jnp %
        sp[row] = sinf(pn);
        cp[row] = cosf(pn);
    }
}

// ---------------------------------------------------------------------------
// K2: per-faction means of sin_p / cos_p (8 factions x 1024 cells)
// ---------------------------------------------------------------------------
__global__ __launch_bounds__(256) void faction_kernel(
    const float* __restrict__ sp, const float* __restrict__ cp,
    float* __restrict__ fm_s, float* __restrict__ fm_c)
{
    __shared__ float rs[256], rc[256];
    const int f = blockIdx.x, t = threadIdx.x;
    float ss = 0.f, sc = 0.f;
    for (int cdx = t; cdx < 1024; cdx += 256) {
        ss += sp[f * 1024 + cdx];
        sc += cp[f * 1024 + cdx];
    }
    rs[t] = ss; rc[t] = sc; __syncthreads();
    for (int off = 128; off > 0; off >>= 1) {
        if (t < off) { rs[t] += rs[t + off]; rc[t] += rc[t + off]; }
        __syncthreads();
    }
    if (t == 0) { fm_s[f] = rs[0] * (1.f / 1024.f); fm_c[f] = rc[0] * (1.f / 1024.f); }
}

// ---------------------------------------------------------------------------
// K3: global stats (r, tension, mean phase), faction-sync+debate coefficients
// (a_i, b_i), softmax weights.
// ---------------------------------------------------------------------------
__global__ __launch_bounds__(1024) void stats_kernel(
    const float* __restrict__ sp, const float* __restrict__ cp,
    const float* __restrict__ fm_s, const float* __restrict__ fm_c,
    const int* __restrict__ stepPtr,
    float* __restrict__ acoef, float* __restrict__ bcoef,
    float* __restrict__ wgt, float* __restrict__ tension_out)
{
    __shared__ float sh_ms, sh_mc, sh_cmp, sh_smp, sh_invZ;
    __shared__ float red[1024];
    const int t = threadIdx.x;
    if (t == 0) {
        float ms = 0.f, mc = 0.f;
        for (int f = 0; f < 8; ++f) { ms += fm_s[f]; mc += fm_c[f]; }
        ms *= 0.125f; mc *= 0.125f;
        const float r = sqrtf(ms * ms + mc * mc);
        tension_out[0] = fabsf(r - 0.5f) * 2.0f;
        sh_ms = ms; sh_mc = mc;
        if (r > 0.f) { sh_cmp = mc / r; sh_smp = ms / r; }
        else         { sh_cmp = 1.f;    sh_smp = 0.f; }   // atan2(0,0)=0
    }
    __syncthreads();
    const float ms = sh_ms, mc = sh_mc, cmp = sh_cmp, smp = sh_smp;
    const bool debate = (stepPtr[0] > 5);

    float zpart = 0.f;
    for (int i = t; i < NCELLS; i += 1024) {
        const int f = i >> 10;
        float a = 0.85f * sp[i] + 0.15f * fm_s[f];
        float b = 0.85f * cp[i] + 0.15f * fm_c[f];
        if (debate && (i & 1023) < 256) {   // dc = FS/4 = 256; global opinion == (ms, mc)
            a = 0.85f * a + 0.15f * ms;
            b = 0.85f * b + 0.15f * mc;
        }
        acoef[i] = a; bcoef[i] = b;
        const float coh = cp[i] * cmp + sp[i] * smp;   // cos(pn - mean_phase)
        const float e = expf(5.f * coh);
        wgt[i] = e;
        zpart += e;
    }
    red[t] = zpart; __syncthreads();
    for (int off = 512; off > 0; off >>= 1) {
        if (t < off) red[t] += red[t + off];
        __syncthreads();
    }
    if (t == 0) sh_invZ = 1.f / red[0];
    __syncthreads();
    const float invZ = sh_invZ;
    for (int i = t; i < NCELLS; i += 1024) wgt[i] *= invZ;
}

// ---------------------------------------------------------------------------
// K5: tvec partials: part[ic][k] = sum_{i in chunk ic} w_i * tanh(a_i u0[k] + b_i u1[k] + u2[k])
// grid 256 = 16 col-groups x 16 i-chunks
// ---------------------------------------------------------------------------
__global__ __launch_bounds__(256) void tanhgrid_kernel(
    const float* __restrict__ U,
    const float* __restrict__ acoef, const float* __restrict__ bcoef,
    const float* __restrict__ wgt, float* __restrict__ part)
{
    const int kg = blockIdx.x & 15;
    const int ic = blockIdx.x >> 4;
    const int t = threadIdx.x;
    const int k = kg * 64 + (t & 63);
    const int isub = t >> 6;
    const float u0 = U[k], u1 = U[1024 + k], u2 = U[2048 + k];
    float acc = 0.f;
    const int i0 = ic * 512 + isub * 128;
    for (int i = i0; i < i0 + 128; ++i) {
        const float z = fmaf(acoef[i], u0, fmaf(bcoef[i], u1, u2));
        acc = fmaf(wgt[i], tanhf(z), acc);
    }
    __shared__ float sm[256];
    sm[t] = acc; __syncthreads();
    if (t < 64)
        part[ic * 1024 + k] = sm[t] + sm[t + 64] + sm[t + 128] + sm[t + 192];
}

// K6: tpad row 0 = tvec = sum_ic part[ic][k]; zero tpad rows 1..15
__global__ __launch_bounds__(256) void colreduce_kernel(
    const float* __restrict__ part, float* __restrict__ tpad)
{
    const int k = blockIdx.x * 256 + threadIdx.x;
    float s = 0.f;
    for (int ic = 0; ic < 16; ++ic) s += part[ic * 1024 + k];
    tpad[k] = s;
    for (int r = 1; r < 16; ++r) tpad[r * 1024 + k] = 0.f;
}

// ---------------------------------------------------------------------------
extern "C" void kernel_launch(void* const* d_in, const int* in_sizes, int n_in,
                              void* d_out, int out_size, void* d_ws, size_t ws_size,
                              hipStream_t stream)
{
    (void)in_sizes; (void)n_in; (void)out_size; (void)ws_size;
    const float* x        = (const float*)d_in[0];
    const float* phases   = (const float*)d_in[1];
    const float* omegas   = (const float*)d_in[2];
    const float* coupling = (const float*)d_in[3];
    const float* Kc       = (const float*)d_in[4];
    const float* w_p2h    = (const float*)d_in[5];
    const float* b_p2h    = (const float*)d_in[6];
    const float* w_enc1   = (const float*)d_in[7];
    const float* b_enc1   = (const float*)d_in[8];
    const float* w_enc2   = (const float*)d_in[9];
    const float* b_enc2   = (const float*)d_in[10];
    const float* w_out    = (const float*)d_in[11];
    const float* b_out    = (const float*)d_in[12];
    const int*   step     = (const int*)d_in[13];

    float* out = (float*)d_out;        // [0..511]=output, [512]=tension
    float* ws  = (float*)d_ws;

    float* sin_o = ws;                 // 8192
    float* cos_o = sin_o + 8192;       // 8192
    float* sp    = cos_o + 8192;       // 8192
    float* cp    = sp + 8192;          // 8192
    float* wgt   = cp + 8192;          // 8192
    float* acoef = wgt + 8192;         // 8192
    float* bcoef = acoef + 8192;       // 8192
    float* fm_s  = bcoef + 8192;       // 8
    float* fm_c  = fm_s + 8;           // 8
    float* scal  = fm_c + 8;           // 16
    float* Ppad  = scal + 16;          // 16*1536 (rows 3..15 zero)
    float* U     = Ppad + 16 * 1536;   // 3*1024
    float* tpad  = U + 3 * 1024;       // 16*1024 (rows 1..15 zero)
    float* wpad  = tpad + 16 * 1024;   // 16*1024 (rows 1..15 zero)
    float* part  = wpad + 16 * 1024;   // 16*1024

    prep_kernel<<<1, 256, 0, stream>>>(x, phases, w_p2h, b_p2h,
                                       sin_o, cos_o, scal, Ppad, wpad);
    // U = Ppad @ W1 (+ b_enc1 into row 2): cell_hiddens@W1_top + x@W1_bot + b1
    rank_gemm_wmma<<<64, 256, 0, stream>>>(Ppad, 1536, w_enc1, 1024, b_enc1, 2, U, 3);
    // HBM-bound 268 MB stream: sigmoid-weighted row sums -> new phases -> sin/cos
    kuramoto_rows<<<1024, 256, 0, stream>>>(coupling, phases, omegas, Kc,
                                            sin_o, cos_o, scal, sp, cp);
    faction_kernel<<<8, 256, 0, stream>>>(sp, cp, fm_s, fm_c);
    stats_kernel<<<1, 1024, 0, stream>>>(sp, cp, fm_s, fm_c, step,
                                         acoef, bcoef, wgt, out + 512);
    tanhgrid_kernel<<<256, 256, 0, stream>>>(U, acoef, bcoef, wgt, part);
    colreduce_kernel<<<4, 256, 0, stream>>>(part, tpad);
    // weighted = tvec @ W2 + b2 -> wpad row 0   (softmax weights sum to 1)
    rank_gemm_wmma<<<64, 256, 0, stream>>>(tpad, 1024, w_enc2, 1024, b_enc2, 0, wpad, 1);
    // output = weighted @ W_out + b_out -> d_out[0..511]
    rank_gemm_wmma<<<32, 256, 0, stream>>>(wpad, 1024, w_out, 512, b_out, 0, out, 1);
}